// F0ProcessorCell_74165495267418
// MI455X (gfx1250) — compile-verified
//
#include <hip/hip_runtime.h>

// F0ProcessorCell scan, reset-parallelized 3-pass formulation for gfx1250.
// B=2048 rows x T=16000 steps, f32. Chunked into 64-element chunks; any
// element with mn >= 1.0 fully resets the carry, making chunk tails exact
// without knowing the entry carry. Pass2 stitches carries; pass3 fixes
// chunk prefixes.

#define T_LEN    16000
#define CHUNK    64
#define NCHUNK   (T_LEN / CHUNK)   // 250 chunks per row
#define BLK1     128
#define LDS_STRIDE 65              // 64 floats + 1 pad -> conflict-free ds reads

// Exactly mirrors the reference step's f32 arithmetic (no FMA contraction).
__device__ __forceinline__ void f0_step(float m, float thr, float& note, float& steps) {
    float na     = fminf(fmaxf(m, 0.0f), 1.0f);                       // sat_relu(mn, 0)
    float re     = fminf(fmaxf(__fsub_rn(steps, thr), 0.0f), 1.0f);   // sat_relu(prev_steps, thr)
    float one_na = __fsub_rn(1.0f, na);
    float one_re = __fsub_rn(1.0f, re);
    float t1     = __fmul_rn(na, m);
    float t2     = __fmul_rn(__fmul_rn(one_na, note), one_re);
    float new_note  = __fadd_rn(t1, t2);
    float new_steps = __fmul_rn(__fmul_rn(__fadd_rn(steps, 1.0f), one_na), one_re);
    note  = new_note;
    steps = new_steps;
}

// ---------------- Pass 1: per-chunk scan from dummy carry ----------------
// One thread per (row, chunk). Stages the chunk into LDS via the gfx1250
// async global->LDS path, then runs the dependent 64-step scan out of LDS.
__global__ __launch_bounds__(BLK1) void f0_pass1(
    const float* __restrict__ mn, const float* __restrict__ rel,
    float* __restrict__ out, float4* __restrict__ rec, int total)
{
    __shared__ float lds[BLK1 * LDS_STRIDE];   // 33,280 bytes

    int g = blockIdx.x * BLK1 + threadIdx.x;
    if (g >= total) return;
    int r = g / NCHUNK;
    int c = g - r * NCHUNK;
    const float* src = mn  + (size_t)r * T_LEN + (size_t)c * CHUNK;
    float*       dst = out + (size_t)r * T_LEN + (size_t)c * CHUNK;

    float* myLds = &lds[threadIdx.x * LDS_STRIDE];
    unsigned ldsoff = (unsigned)(size_t)myLds;   // low 32 bits of generic addr == LDS byte offset

    // 16 x async b128: copy this thread's 256B chunk into its LDS region.
    // Tracked by ASYNCcnt (CDNA5 async memory path).
    #pragma unroll
    for (int k = 0; k < CHUNK / 4; ++k) {
        const float* ga = src + k * 4;
        asm volatile("global_load_async_to_lds_b128 %0, %1, off"
                     :: "v"(ldsoff + (unsigned)(k * 16)), "v"(ga)
                     : "memory");
    }

    float thr = __fmul_rn(rel[0], 250.0f);       // release_duration * FRAME_RATE

    asm volatile("s_wait_asynccnt 0" ::: "memory");

    float note = 0.0f, steps = 0.0f;
    int firstReset = CHUNK;                      // CHUNK == "no reset in chunk"
    for (int t = 0; t < CHUNK; t += 4) {
        float4 ob;
        #pragma unroll
        for (int j = 0; j < 4; ++j) {
            float m = myLds[t + j];
            if (firstReset == CHUNK && m >= 1.0f) firstReset = t + j;
            f0_step(m, thr, note, steps);
            (&ob.x)[j] = note;
        }
        *(float4*)(dst + t) = ob;                // prefix (< firstReset) fixed by pass 3
    }
    rec[g] = make_float4(note, steps, __int_as_float(firstReset), 0.0f);
}

// ---------------- Pass 2: stitch carries across chunks (per row) ----------------
__global__ __launch_bounds__(256) void f0_pass2(
    const float* __restrict__ mn, const float* __restrict__ rel,
    const float4* __restrict__ rec, float2* __restrict__ entry, int B)
{
    int r = blockIdx.x * 256 + threadIdx.x;
    if (r >= B) return;
    float thr = __fmul_rn(rel[0], 250.0f);
    float note = 0.0f, steps = 0.0f;
    const float4* rrec = rec   + (size_t)r * NCHUNK;
    float2*       rent = entry + (size_t)r * NCHUNK;
    for (int c = 0; c < NCHUNK; ++c) {
        rent[c] = make_float2(note, steps);
        float4 rc = rrec[c];
        int fr = __float_as_int(rc.z);
        if (fr < CHUNK) {                        // chunk had a reset: recorded exit is exact
            note = rc.x; steps = rc.y;
        } else {                                 // rare: rescan with true entry carry
            const float* src = mn + (size_t)r * T_LEN + (size_t)c * CHUNK;
            for (int t = 0; t < CHUNK; ++t) f0_step(src[t], thr, note, steps);
        }
    }
}

// ---------------- Pass 3: fix chunk prefixes with true entry carries ----------------
__global__ __launch_bounds__(256) void f0_pass3(
    const float* __restrict__ mn, const float* __restrict__ rel,
    const float4* __restrict__ rec, const float2* __restrict__ entry,
    float* __restrict__ out, int total)
{
    int g = blockIdx.x * 256 + threadIdx.x;
    if (g >= total) return;
    float4 rc = rec[g];
    int fr = __float_as_int(rc.z);
    if (fr <= 0) return;                         // reset at t=0: pass1 output already exact
    if (fr > CHUNK) fr = CHUNK;
    int r = g / NCHUNK;
    int c = g - r * NCHUNK;
    float2 e = entry[g];
    float note = e.x, steps = e.y;
    float thr = __fmul_rn(rel[0], 250.0f);
    const float* src = mn  + (size_t)r * T_LEN + (size_t)c * CHUNK;
    float*       dst = out + (size_t)r * T_LEN + (size_t)c * CHUNK;
    for (int t = 0; t < fr; ++t) {
        f0_step(src[t], thr, note, steps);
        dst[t] = note;
    }
}

extern "C" void kernel_launch(void* const* d_in, const int* in_sizes, int n_in,
                              void* d_out, int out_size, void* d_ws, size_t ws_size,
                              hipStream_t stream)
{
    (void)n_in; (void)out_size; (void)ws_size;
    const float* mn  = (const float*)d_in[0];
    const float* rel = (const float*)d_in[1];   // scalar release_duration
    float* out = (float*)d_out;

    int B     = in_sizes[0] / T_LEN;            // 2048
    int total = B * NCHUNK;                     // 512000 chunks

    // Workspace: [total] float4 chunk records, then [total] float2 entry carries (~12.3 MB).
    float4* rec   = (float4*)d_ws;
    float2* entry = (float2*)((char*)d_ws + (size_t)total * sizeof(float4));

    hipLaunchKernelGGL(f0_pass1, dim3((total + BLK1 - 1) / BLK1), dim3(BLK1), 0, stream,
                       mn, rel, out, rec, total);
    hipLaunchKernelGGL(f0_pass2, dim3((B + 255) / 256), dim3(256), 0, stream,
                       mn, rel, rec, entry, B);
    hipLaunchKernelGGL(f0_pass3, dim3((total + 255) / 256), dim3(256), 0, stream,
                       mn, rel, rec, entry, out, total);
}